// FeatureMap_37237366456700
// MI455X (gfx1250) — compile-verified
//
#include <hip/hip_runtime.h>
#include <hip/hip_bf16.h>

typedef __attribute__((ext_vector_type(2))) float v2f;
typedef __attribute__((ext_vector_type(8))) float v8f;

#define NCLS   24
#define BATCH  8
#define CHANS  128
#define WLO    128
#define NPIX   (128*128)      // low-res pixels per batch
#define WHI    512
#define KPAD   32             // class planes allocated (24 written; 24..31 junk, never used)
#define KCHUNKS 64
#define KLEN   (NPIX / KCHUNKS)

// ---------------- zero accumulators ----------------
__global__ void fm_zero(float* __restrict__ p, int n) {
    int i = blockIdx.x * blockDim.x + threadIdx.x;
    if (i < n) p[i] = 0.0f;
}

// ------- labels -> per-class pixel planes (exact fp32 counts), class-major -------
__global__ void fm_hist(const long long* __restrict__ label,
                        float* __restrict__ hist,
                        float* __restrict__ counts) {
    __shared__ float scnt[NCLS];
    const int t = threadIdx.x;
    if (t < NCLS) scnt[t] = 0.0f;
    __syncthreads();

    const int b = blockIdx.y;
    const int p = blockIdx.x * blockDim.x + t;       // low-res pixel id
    const int i = p >> 7;
    const int j = p & (WLO - 1);

    const long long* lb = label + (size_t)b * WHI * WHI
                                + (size_t)(i * 4) * WHI + (size_t)(j * 4);
    int lab[16];
#pragma unroll
    for (int r = 0; r < 4; ++r) {                    // 4 rows x 4 int64 = 32B rows
        const long long* row = lb + (size_t)r * WHI;
        lab[r * 4 + 0] = (int)row[0];
        lab[r * 4 + 1] = (int)row[1];
        lab[r * 4 + 2] = (int)row[2];
        lab[r * 4 + 3] = (int)row[3];
    }

    float* hb = hist + (size_t)b * KPAD * NPIX + p;
#pragma unroll
    for (int k = 0; k < NCLS; ++k) {
        int cnt = 0;
#pragma unroll
        for (int l = 0; l < 16; ++l) cnt += (lab[l] == k) ? 1 : 0;
        hb[(size_t)k * NPIX] = (float)cnt;           // coalesced across the warp
        if (cnt) atomicAdd(&scnt[k], (float)cnt);    // label==-1 never counted
    }

    __syncthreads();
    if (t < NCLS) atomicAdd(&counts[b * NCLS + t], scnt[t]);
}

// ---- fp32 WMMA GEMM: sums[b,k,c] += data[b,c,p] * hist[b,k,p]  (2 M-tiles/wave) ----
__global__ void fm_gemm(const float* __restrict__ data,
                        const float* __restrict__ hist,
                        float* __restrict__ sums) {
    const int lane = threadIdx.x;        // one wave32 per block
    const int hl   = lane >> 4;          // half-wave selects K pair (ISA 16x4 layout)
    const int lm   = lane & 15;          // M row (A) / N column (B)
    const int kc   = blockIdx.x;
    const int mp   = blockIdx.y;         // pair of 16-channel M tiles (32 channels)
    const int b    = blockIdx.z;

    // A fragments: row = channel (M), col = pixel (K); lane lm = M, hl picks K pair
    const float* d0 = data + ((size_t)(b * CHANS + mp * 32 + lm)) * NPIX + 2 * hl;
    const float* d1 = d0 + (size_t)16 * NPIX;
    // B fragments: class-major planes; lane lm = N column, contiguous K -> float2
    const float* hb = hist + (size_t)b * KPAD * NPIX + 2 * hl;
    const float* h0 = hb + (size_t)lm * NPIX;          // classes 0..15
    const float* h1 = hb + (size_t)(16 + lm) * NPIX;   // classes 16..23 (+junk cols)

    v8f acc00 = {}, acc01 = {}, acc10 = {}, acc11 = {};

    int p0 = kc * KLEN;
    const int pend = p0 + KLEN;
    for (; p0 < pend; p0 += 4) {
        __builtin_prefetch(d0 + p0 + 128, 0, 3);               // global_prefetch_b8
        __builtin_prefetch(d1 + p0 + 128, 0, 3);
        v2f a0 = *(const v2f*)(d0 + p0);
        v2f a1 = *(const v2f*)(d1 + p0);
        v2f b0 = *(const v2f*)(h0 + p0);
        v2f b1 = *(const v2f*)(h1 + p0);
        acc00 = __builtin_amdgcn_wmma_f32_16x16x4_f32(false, a0, false, b0,
                                                      (short)0, acc00, false, false);
        acc01 = __builtin_amdgcn_wmma_f32_16x16x4_f32(false, a0, false, b1,
                                                      (short)0, acc01, false, false);
        acc10 = __builtin_amdgcn_wmma_f32_16x16x4_f32(false, a1, false, b0,
                                                      (short)0, acc10, false, false);
        acc11 = __builtin_amdgcn_wmma_f32_16x16x4_f32(false, a1, false, b1,
                                                      (short)0, acc11, false, false);
    }

    // D layout: VGPR v -> M = v + 8*hl (channel), N = lm (class)
    const size_t srow0 = ((size_t)b * NCLS + lm) * CHANS + mp * 32;
    const size_t srow1 = ((size_t)b * NCLS + 16 + lm) * CHANS + mp * 32;
#pragma unroll
    for (int v = 0; v < 8; ++v) {
        const int m = v + 8 * hl;
        atomicAdd(&sums[srow0 + m],      acc00[v]);
        atomicAdd(&sums[srow0 + 16 + m], acc10[v]);
        if (lm < 8) {                                   // only classes 16..23 are real
            atomicAdd(&sums[srow1 + m],      acc01[v]);
            atomicAdd(&sums[srow1 + 16 + m], acc11[v]);
        }
    }
}

// ---------------- divide + mask -> d_out ----------------
__global__ void fm_final(const float* __restrict__ sums,
                         const float* __restrict__ counts,
                         float* __restrict__ out) {
    const int idx = blockIdx.x * blockDim.x + threadIdx.x;
    const int NF = BATCH * CHANS * NCLS;
    if (idx < NF) {
        const int b = idx / (CHANS * NCLS);
        const int r = idx % (CHANS * NCLS);
        const int c = r / NCLS;
        const int k = r % NCLS;
        const float cnt = counts[b * NCLS + k];
        out[idx] = sums[((size_t)b * NCLS + k) * CHANS + c] / (cnt + 1e-8f);
    } else if (idx < NF + BATCH * NCLS) {
        const int j = idx - NF;
        out[idx] = (counts[j] > 0.0f) ? 1.0f : 0.0f;
    }
}

extern "C" void kernel_launch(void* const* d_in, const int* in_sizes, int n_in,
                              void* d_out, int out_size, void* d_ws, size_t ws_size,
                              hipStream_t stream) {
    const float*     data  = (const float*)d_in[0];       // (8,128,128,128) fp32
    const long long* label = (const long long*)d_in[1];   // (8,512,512) int64
    float* out = (float*)d_out;                           // 24576 feature + 192 mask

    float* sums   = (float*)d_ws;                         // 8*24*128
    float* counts = sums + BATCH * NCLS * CHANS;          // 8*24
    float* hist   = counts + BATCH * NCLS;                // 8*32*16384 fp32 (~16 MB)

    const int nacc = BATCH * NCLS * CHANS + BATCH * NCLS;
    fm_zero<<<(nacc + 255) / 256, 256, 0, stream>>>(sums, nacc);

    dim3 hg(NPIX / 256, BATCH);
    fm_hist<<<hg, 256, 0, stream>>>(label, hist, counts);

    dim3 gg(KCHUNKS, CHANS / 32, BATCH);
    fm_gemm<<<gg, 32, 0, stream>>>(data, hist, sums);

    const int nf = BATCH * CHANS * NCLS + BATCH * NCLS;
    fm_final<<<(nf + 255) / 256, 256, 0, stream>>>(sums, counts, out);
}